// DynamicGraphConv_54966991454711
// MI455X (gfx1250) — compile-verified
//
#include <hip/hip_runtime.h>
#include <hip/hip_bf16.h>
#include <math.h>

typedef __attribute__((ext_vector_type(2))) float v2f;
typedef __attribute__((ext_vector_type(8))) float v8f;

#define N_NODES 50000
#define N_EDGES 800000
#define DIM     128

// ---------------- Kernel 1: zero the h accumulator (in d_ws) ----------------
__global__ void dgc_zero_h(float4* __restrict__ h4, int n4) {
    int i = blockIdx.x * blockDim.x + threadIdx.x;
    if (i < n4) h4[i] = make_float4(0.f, 0.f, 0.f, 0.f);
}

// ---------------- Kernel 2: edge scatter  h[dst] += w * x[src] --------------
// One wave32 per edge. Each lane handles a float4 (lane*16B) => 512B row.
// Edge metadata is wave-uniform (derived from wave id) -> scalar loads.
__global__ void dgc_edge_scatter(const float* __restrict__ x,
                                 const int*   __restrict__ esrc,
                                 const int*   __restrict__ edst,
                                 const float* __restrict__ ew,
                                 float*       __restrict__ h) {
    int wave = (blockIdx.x * blockDim.x + threadIdx.x) >> 5;
    int lane = threadIdx.x & 31;
    if (wave >= N_EDGES) return;

    int   s = esrc[wave];
    int   d = edst[wave];
    float w = ew[wave];

    const float4* xrow = (const float4*)(x + (size_t)s * DIM);
    float4 v = xrow[lane];

    float* hp = h + (size_t)d * DIM + lane * 4;
    atomicAdd(hp + 0, w * v.x);
    atomicAdd(hp + 1, w * v.y);
    atomicAdd(hp + 2, w * v.z);
    atomicAdd(hp + 3, w * v.w);
}

// ---------------- Kernel 3: WMMA gate mat-vec + blend -----------------------
// One wave32 per 16-node tile. D = A(16x4 of h) x B(4x16 of replicated gate_w)
// accumulated over K=128 via 32x v_wmma_f32_16x16x4_f32. Every column of D is
// identical, so D[m][*] = dot(h[m], gate_w).
// f32 16x16x4 operand layout (ISA 7.12.2): K = (lane<16 ? 0 : 2) + vgpr_idx
// for both A (M=lane%16) and B (N=lane%16, value independent of N here).
__global__ void dgc_gate_blend(const float* __restrict__ x,
                               const float* __restrict__ h,
                               const float* __restrict__ gw,
                               const float* __restrict__ gb,
                               float*       __restrict__ out) {
    int wave = (blockIdx.x * blockDim.x + threadIdx.x) >> 5;
    int lane = threadIdx.x & 31;
    const int ntiles = N_NODES / 16;   // 3125, exact
    if (wave >= ntiles) return;

    int m    = lane & 15;
    int koff = (lane >> 4) << 1;       // 0 for lanes 0-15, 2 for lanes 16-31
    const float* hrow = h + (size_t)(wave * 16 + m) * DIM;

    v8f c = {};
#pragma unroll
    for (int k0 = 0; k0 < DIM; k0 += 4) {
        int kk = k0 + koff;
        v2f a; a.x = hrow[kk]; a.y = hrow[kk + 1];
        v2f b; b.x = gw[kk];   b.y = gw[kk + 1];
        // 8 args: (neg_a, A, neg_b, B, c_mod, C, reuse_a, reuse_b)
        c = __builtin_amdgcn_wmma_f32_16x16x4_f32(
                false, a, false, b, (short)0, c, false, false);
    }

    float bias = gb[0];

    // lanes 0-15 hold dots M=0..7 in c[0..7]; lanes 16-31 hold M=8..15.
    float dot[16];
#pragma unroll
    for (int i = 0; i < 8; ++i) {
        dot[i]     = __shfl(c[i], 0,  32);
        dot[i + 8] = __shfl(c[i], 16, 32);
    }

    const float4* x4 = (const float4*)x;
    const float4* h4 = (const float4*)h;
    float4*       o4 = (float4*)out;
#pragma unroll
    for (int mm = 0; mm < 16; ++mm) {
        int node = wave * 16 + mm;
        float g  = 1.0f / (1.0f + __expf(-(dot[mm] + bias)));
        float og = 1.0f - g;
        size_t idx = (size_t)node * (DIM / 4) + lane;   // 32 float4 per row
        float4 hv = h4[idx];
        float4 xv = x4[idx];
        float4 ov;
        ov.x = g * hv.x + og * xv.x;
        ov.y = g * hv.y + og * xv.y;
        ov.z = g * hv.z + og * xv.z;
        ov.w = g * hv.w + og * xv.w;
        o4[idx] = ov;
    }
}

// ---------------------------------------------------------------------------
extern "C" void kernel_launch(void* const* d_in, const int* in_sizes, int n_in,
                              void* d_out, int out_size, void* d_ws, size_t ws_size,
                              hipStream_t stream) {
    const float* x    = (const float*)d_in[0];
    const int*   esrc = (const int*)  d_in[1];
    const int*   edst = (const int*)  d_in[2];
    const float* ew   = (const float*)d_in[3];
    const float* gw   = (const float*)d_in[4];
    const float* gb   = (const float*)d_in[5];
    float*       out  = (float*)d_out;
    float*       h    = (float*)d_ws;            // 50000*128*4 = 25.6 MB

    // 1) h = 0
    {
        int n4 = N_NODES * DIM / 4;              // 1.6M float4
        int blocks = (n4 + 255) / 256;
        dgc_zero_h<<<blocks, 256, 0, stream>>>((float4*)h, n4);
    }
    // 2) scatter-add: one wave per edge (8 edges per 256-thread block)
    {
        int blocks = (N_EDGES + 7) / 8;          // 100000 blocks
        dgc_edge_scatter<<<blocks, 256, 0, stream>>>(x, esrc, edst, ew, h);
    }
    // 3) WMMA gate + blend: one wave per 16 nodes
    {
        int waves  = N_NODES / 16;               // 3125
        int blocks = (waves * 32 + 255) / 256;   // 391 blocks
        dgc_gate_blend<<<blocks, 256, 0, stream>>>(x, h, gw, gb, out);
    }
}